// RelNwRegr_17463337025833
// MI455X (gfx1250) — compile-verified
//
#include <hip/hip_runtime.h>
#include <hip/hip_bf16.h>

// Relational NW regression, fused single-pass over r (the 64MB dominant stream).
// f32 WMMA (V_WMMA_F32_16X16X4_F32) chained x8 for the K=32 GEMM.
// Memory roofline: 64MB of r @ 23.3 TB/s ~= 2.9us; JSPLIT=16 gives ~8 waves/SIMD
// so the HBM stream stays saturated.

typedef __attribute__((ext_vector_type(2))) float v2f;
typedef __attribute__((ext_vector_type(8))) float v8f;

#define NQ   2048
#define NB   8192
#define DIMK 32
#define JSPLIT 16           // column splits (grid.y) -> 2048 blocks, 8192 waves
#define TILES_PER_BLOCK 32  // 32 * 16 = 512 columns per block
#define TILES_PER_WAVE  8   // 4 waves per block

// workspace layout (floats): [0,8192) b2 | [8192,10240) num | [10240,12288) den
#define WS_B2  0
#define WS_NUM NB
#define WS_DEN (NB + NQ)

__global__ __launch_bounds__(256) void rnw_init(const float* __restrict__ xb,
                                                const float* __restrict__ w,
                                                float* __restrict__ ws) {
    int i = blockIdx.x * blockDim.x + threadIdx.x;
    if (i < NB) {
        const float* xr = xb + i * DIMK;
        float acc = 0.f;
#pragma unroll
        for (int k = 0; k < DIMK; ++k) {
            float v = xr[k] * w[k];
            acc = fmaf(v, v, acc);
        }
        ws[WS_B2 + i] = acc;
    }
    if (i < 2 * NQ) ws[WS_NUM + i] = 0.f;   // zero num+den accumulators
}

__global__ __launch_bounds__(128) void rnw_main(const float* __restrict__ xb,
                                                const float* __restrict__ yb,
                                                const float* __restrict__ xq,
                                                const float* __restrict__ r,
                                                const float* __restrict__ sigma,
                                                const float* __restrict__ r_scale,
                                                const float* __restrict__ w,
                                                float* __restrict__ ws) {
    __shared__ float q2s[16];
    __shared__ float num_l[16];
    __shared__ float den_l[16];

    const int tid   = threadIdx.x;
    const int lane  = tid & 31;
    const int wave  = tid >> 5;
    const int row16 = lane & 15;   // A: M index; B: N index; C: N index
    const int hi    = lane >> 4;   // half-wave selects K pair (A/B) and M half (C)
    const int lo    = hi * 2;
    const int qbase = blockIdx.x * 16;

    // per-block prologue: q2 for the 16 query rows + zero LDS accumulators
    if (tid < 16) {
        const float* xr = xq + (size_t)(qbase + tid) * DIMK;
        float acc = 0.f;
#pragma unroll
        for (int k = 0; k < DIMK; ++k) {
            float v = xr[k] * w[k];
            acc = fmaf(v, v, acc);
        }
        q2s[tid]   = acc;
        num_l[tid] = 0.f;
        den_l[tid] = 0.f;
    }
    __syncthreads();

    // Preload A tile (16x32 query block, weighted) per the 32-bit A-matrix
    // layout: lane row16 holds row M=row16; lanes 16-31 hold K-pairs offset +2.
    v2f a[8], w2[8];
    {
        const float* xr = xq + (size_t)(qbase + row16) * DIMK;
#pragma unroll
        for (int s = 0; s < 8; ++s) {
            int k0 = 4 * s + lo;
            w2[s].x = w[k0];
            w2[s].y = w[k0 + 1];
            a[s].x = xr[k0]     * w2[s].x;
            a[s].y = xr[k0 + 1] * w2[s].y;
        }
    }
    float q2r[8];
#pragma unroll
    for (int v = 0; v < 8; ++v) q2r[v] = q2s[v + 8 * hi];  // C row M = v + 8*hi

    // fold sigma / r_scale so the kernel exp is a single v_exp_f32 (2^x)
    const float LOG2E = 1.4426950408889634f;
    const float c1 = -LOG2E / sigma[0];
    const float c2 =  LOG2E * r_scale[0];

    float num_acc[8], den_acc[8];
#pragma unroll
    for (int v = 0; v < 8; ++v) { num_acc[v] = 0.f; den_acc[v] = 0.f; }

    const int t0 = blockIdx.y * TILES_PER_BLOCK + wave * TILES_PER_WAVE;
    for (int t = t0; t < t0 + TILES_PER_WAVE; ++t) {
        const int jbase = t * 16;
        const float b2n = ws[WS_B2 + jbase + row16];  // per-column norms
        const float yn  = yb[jbase + row16];
        const float* xbr = xb + (size_t)(jbase + row16) * DIMK;

        // S = A (16x32) @ B (32x16) via 8 chained f32 WMMAs (K=4 each)
        v8f c = {};
#pragma unroll
        for (int s = 0; s < 8; ++s) {
            int k0 = 4 * s + lo;
            v2f b;
            b.x = xbr[k0]     * w2[s].x;
            b.y = xbr[k0 + 1] * w2[s].y;
            c = __builtin_amdgcn_wmma_f32_16x16x4_f32(
                    /*neg_a=*/false, a[s], /*neg_b=*/false, b,
                    /*c_mod=*/(short)0, c, /*reuse_a=*/false, /*reuse_b=*/false);
        }

        // fuse: d2 -> dist -> exp kernel -> row-weighted accumulation
#pragma unroll
        for (int v = 0; v < 8; ++v) {
            float rv = r[(size_t)(qbase + v + 8 * hi) * NB + jbase + row16];
            float d2 = q2r[v] + b2n - 2.0f * c[v];
            float dist = __builtin_amdgcn_sqrtf(fmaxf(d2, 0.0f));
            float kv = __builtin_amdgcn_exp2f(fmaf(c2, rv, c1 * dist));
            den_acc[v] += kv;
            num_acc[v] = fmaf(kv, yn, num_acc[v]);
        }
    }

    // reduce across lanes/waves: LDS float atomics (ds_add_f32)
#pragma unroll
    for (int v = 0; v < 8; ++v) {
        atomicAdd(&num_l[v + 8 * hi], num_acc[v]);
        atomicAdd(&den_l[v + 8 * hi], den_acc[v]);
    }
    __syncthreads();

    // one global_atomic_add_f32 per row per block (column splits combine here)
    if (tid < 16) {
        atomicAdd(&ws[WS_NUM + qbase + tid], num_l[tid]);
        atomicAdd(&ws[WS_DEN + qbase + tid], den_l[tid]);
    }
}

__global__ __launch_bounds__(256) void rnw_fin(const float* __restrict__ ws,
                                               float* __restrict__ out) {
    int i = blockIdx.x * blockDim.x + threadIdx.x;
    if (i < NQ) out[i] = ws[WS_NUM + i] / (ws[WS_DEN + i] + 1e-8f);
}

extern "C" void kernel_launch(void* const* d_in, const int* in_sizes, int n_in,
                              void* d_out, int out_size, void* d_ws, size_t ws_size,
                              hipStream_t stream) {
    (void)in_sizes; (void)n_in; (void)out_size; (void)ws_size;
    const float* xb      = (const float*)d_in[0];  // (8192, 32)
    const float* yb      = (const float*)d_in[1];  // (8192,)
    const float* xq      = (const float*)d_in[2];  // (2048, 32)
    const float* r       = (const float*)d_in[3];  // (2048, 8192)
    const float* sigma   = (const float*)d_in[4];  // (1,)
    const float* r_scale = (const float*)d_in[5];  // (1,)
    const float* w       = (const float*)d_in[6];  // (32,)
    float* out = (float*)d_out;
    float* ws  = (float*)d_ws;

    // 1) b2 precompute + zero accumulators
    rnw_init<<<dim3((NB + 255) / 256), dim3(256), 0, stream>>>(xb, w, ws);
    // 2) fused WMMA + kernel + reduction, streaming r exactly once
    rnw_main<<<dim3(NQ / 16, JSPLIT), dim3(128), 0, stream>>>(
        xb, yb, xq, r, sigma, r_scale, w, ws);
    // 3) normalize
    rnw_fin<<<dim3((NQ + 255) / 256), dim3(256), 0, stream>>>(ws, out);
}